// ResidualFSQ_55714315763968
// MI455X (gfx1250) — compile-verified
//
#include <hip/hip_runtime.h>
#include <math.h>

typedef float v2f __attribute__((ext_vector_type(2)));
typedef float v8f __attribute__((ext_vector_type(8)));

#define DDIM 512
#define NQ   8
#define TOK_PER_WAVE 16
#define WAVES 8
#define THREADS 256
#define TOK_PER_BLOCK (WAVES * TOK_PER_WAVE)

__global__ __launch_bounds__(THREADS)
void fsq_fused_kernel(const float* __restrict__ x,
                      const float* __restrict__ Win,
                      const float* __restrict__ bin,
                      const float* __restrict__ Wout,
                      const float* __restrict__ bout,
                      float* __restrict__ out,
                      float* __restrict__ idxOut)
{
    __shared__ float  woutsh[DDIM * 4];          // 8 KB: W_out TRANSPOSED [d][c] -> b64 B frags
    __shared__ float  boutsh[DDIM];              // 2 KB: b_out
    __shared__ float4 qsh[WAVES][16];            // 2 KB: quantized 4-vectors per token
    __shared__ float  idxsh[WAVES][16][NQ];      // 4 KB: indices staged for coalesced store

    const int tid  = threadIdx.x;
    const int lane = tid & 31;
    const int wave = tid >> 5;
    const int tokBase = blockIdx.x * TOK_PER_BLOCK + wave * TOK_PER_WAVE;

    // ---- stage W_out (4x512) transposed into LDS: woutsh[d*4+c] = Wout[c*512+d] ----
    {
        const float4* w4 = (const float4*)Wout;  // 512 float4; idx -> c = idx>>7, d0 = (idx&127)*4
        #pragma unroll
        for (int rep = 0; rep < 2; ++rep) {
            int idx = tid + rep * 256;
            float4 w = w4[idx];
            int c  = idx >> 7;
            int d0 = (idx & 127) * 4;
            woutsh[(d0 + 0) * 4 + c] = w.x;
            woutsh[(d0 + 1) * 4 + c] = w.y;
            woutsh[(d0 + 2) * 4 + c] = w.z;
            woutsh[(d0 + 3) * 4 + c] = w.w;
        }
    }
    if (tid < DDIM / 4) ((float4*)boutsh)[tid] = ((const float4*)bout)[tid];

    // ---- preload this lane's W_in rows into registers (reused by all 16 tokens) ----
    // lane handles d-rows (i*32+lane)*4 + j  for i=0..3, j=0..3  (W_in is (512,4) row-major)
    float4 wf[16];
    {
        const float4* win4 = (const float4*)Win;  // 512 float4, one per d-row
        #pragma unroll
        for (int i = 0; i < 4; ++i)
            #pragma unroll
            for (int j = 0; j < 4; ++j)
                wf[i * 4 + j] = win4[(i * 32 + lane) * 4 + j];
    }

    // ---- Phase A: project_in. Double-buffered coalesced b128 streaming + butterfly reduce.
    // After the loop, lane t (t<16) holds z[token t] in z0..z3.
    float z0 = 0.f, z1 = 0.f, z2 = 0.f, z3 = 0.f;
    float4 cur[4];
    {
        const float4* r4 = (const float4*)(x + (size_t)tokBase * DDIM);
        #pragma unroll
        for (int i = 0; i < 4; ++i) cur[i] = r4[i * 32 + lane];
    }
    for (int t = 0; t < TOK_PER_WAVE; ++t) {
        float4 nxt[4];
        if (t < TOK_PER_WAVE - 1) {               // prefetch next token row (hides latency)
            const float4* r4 = (const float4*)(x + (size_t)(tokBase + t + 1) * DDIM);
            #pragma unroll
            for (int i = 0; i < 4; ++i) nxt[i] = r4[i * 32 + lane];
        }
        float4 acc = make_float4(0.f, 0.f, 0.f, 0.f);
        #pragma unroll
        for (int i = 0; i < 4; ++i) {
            float4 v = cur[i];
            const float4 w0 = wf[4 * i + 0], w1 = wf[4 * i + 1];
            const float4 w2 = wf[4 * i + 2], w3 = wf[4 * i + 3];
            acc.x += v.x * w0.x + v.y * w1.x + v.z * w2.x + v.w * w3.x;
            acc.y += v.x * w0.y + v.y * w1.y + v.z * w2.y + v.w * w3.y;
            acc.z += v.x * w0.z + v.y * w1.z + v.z * w2.z + v.w * w3.z;
            acc.w += v.x * w0.w + v.y * w1.w + v.z * w2.w + v.w * w3.w;
        }
        #pragma unroll
        for (int m = 16; m >= 1; m >>= 1) {
            acc.x += __shfl_xor(acc.x, m, 32);
            acc.y += __shfl_xor(acc.y, m, 32);
            acc.z += __shfl_xor(acc.z, m, 32);
            acc.w += __shfl_xor(acc.w, m, 32);
        }
        if (lane == t) { z0 = acc.x; z1 = acc.y; z2 = acc.z; z3 = acc.w; }
        #pragma unroll
        for (int i = 0; i < 4; ++i) cur[i] = nxt[i];
    }

    // ---- Phase B: residual FSQ, lanes 0..15 own one token each (all registers) ----
    if (lane < 16) {
        z0 += bin[0]; z1 += bin[1]; z2 += bin[2]; z3 += bin[3];
        const float hl0 = 3.5f * 1.001f;          // (8-1)*(1+1e-3)/2
        const float hl  = 2.0f * 1.001f;          // (5-1)*(1+1e-3)/2
        const float sh0 = atanhf(0.5f / hl0);     // shift for even level (dim 0)
        float s0 = 1.f, i0 = 1.f;                 // scale / inv-scale for dim0: 7^-q
        float s1 = 1.f, i1 = 1.f;                 // for dims 1..3: 4^-q
        float r0 = z0, r1 = z1, r2 = z2, r3 = z3; // residual
        float q0 = 0.f, q1 = 0.f, q2 = 0.f, q3 = 0.f;
        #pragma unroll
        for (int q = 0; q < NQ; ++q) {
            float rd0 = rintf(tanhf(r0 * i0 + sh0) * hl0 - 0.5f);
            float rd1 = rintf(tanhf(r1 * i1) * hl);
            float rd2 = rintf(tanhf(r2 * i1) * hl);
            float rd3 = rintf(tanhf(r3 * i1) * hl);
            float t0 = rd0 * 0.25f * s0;          // codes/half_width * scale
            float t1 = rd1 * 0.5f  * s1;
            float t2 = rd2 * 0.5f  * s1;
            float t3 = rd3 * 0.5f  * s1;
            r0 -= t0; r1 -= t1; r2 -= t2; r3 -= t3;
            q0 += t0; q1 += t1; q2 += t2; q3 += t3;
            // idx = sum zhat*basis, basis = [1, 8, 40, 200]
            float idx = (rd0 + 4.f) + 8.f * (rd1 + 2.f) + 40.f * (rd2 + 2.f) + 200.f * (rd3 + 2.f);
            idxsh[wave][lane][q] = (float)(int)idx;
            s0 *= (1.f / 7.f); i0 *= 7.f;
            s1 *= 0.25f;       i1 *= 4.f;
        }
        qsh[wave][lane] = make_float4(q0, q1, q2, q3);
    }

    __syncthreads();   // publish woutsh, boutsh, qsh, idxsh

    // ---- coalesced index writeout: 16 tok x 8 = 128 floats per wave ----
    {
        const float4* s = (const float4*)&idxsh[wave][0][0];
        float4* g = (float4*)(idxOut + (size_t)tokBase * NQ);
        g[lane] = s[lane];
    }

    // ---- Phase C: project_out via V_WMMA_F32_16X16X4_F32 (M=16 tok, N=16 cols, K=4) ----
    const int half = lane >> 4;     // 0: K=0,1 | 1: K=2,3 (A/B frag halves)
    const int n    = lane & 15;     // column-in-tile / token-in-tile (A rows)

    // A fragment: lane holds q[token = n][k0], q[n][k0+1], k0 = 2*half (single ds_load_b64)
    v2f a;
    {
        const float2* qp = (const float2*)&qsh[wave][n];
        float2 aa = qp[half];
        a[0] = aa.x; a[1] = aa.y;
    }

    float* const orow = out + (size_t)(tokBase + half * 8) * DDIM + n;
    const float2* const wp = (const float2*)woutsh;   // [d][c] pairs: conflict-free b64
    #pragma unroll 4
    for (int dt = 0; dt < DDIM / 16; ++dt) {
        const int col = dt * 16 + n;
        float2 bb = wp[col * 2 + half];          // {Wout[k0][col], Wout[k0+1][col]}
        v2f b; b[0] = bb.x; b[1] = bb.y;
        const float bias = boutsh[col];
        v8f acc;
        #pragma unroll
        for (int r = 0; r < 8; ++r) acc[r] = bias;   // C = b_out broadcast per column
        acc = __builtin_amdgcn_wmma_f32_16x16x4_f32(
                  false, a, false, b, (short)0, acc, false, false);
        // D frag: VGPR r -> row M = r + 8*half, col n (per ISA C/D layout)
        #pragma unroll
        for (int r = 0; r < 8; ++r)
            orow[(size_t)r * DDIM + dt * 16] = acc[r];
    }
}

extern "C" void kernel_launch(void* const* d_in, const int* in_sizes, int n_in,
                              void* d_out, int out_size, void* d_ws, size_t ws_size,
                              hipStream_t stream) {
    const float* x    = (const float*)d_in[0];
    const float* Win  = (const float*)d_in[1];
    const float* bin  = (const float*)d_in[2];
    const float* Wout = (const float*)d_in[3];
    const float* bout = (const float*)d_in[4];
    float* out = (float*)d_out;

    const int total_tok = in_sizes[0] / DDIM;               // 32768
    float* idxOut = out + (size_t)total_tok * DDIM;         // indices follow `out` flat
    const int blocks = total_tok / TOK_PER_BLOCK;           // 256

    fsq_fused_kernel<<<blocks, THREADS, 0, stream>>>(x, Win, bin, Wout, bout,
                                                     out, idxOut);
}